// CrystallisationManager_9113920602163
// MI455X (gfx1250) — compile-verified
//
#include <hip/hip_runtime.h>

typedef __attribute__((ext_vector_type(2))) float v2f;
typedef __attribute__((ext_vector_type(8))) float v8f;

#define B_    8
#define L_    4096
#define DIM_  1024
#define H_    16
#define M_    64
#define DH_   64
#define NTOK  (B_ * L_)                 // 32768 tokens
#define WAVES_PER_WG 4
#define THREADS_PER_WG (WAVES_PER_WG * 32)
#define TOK_PER_WG (WAVES_PER_WG * 16)  // 64 tokens per workgroup
#define WGS_PER_HEAD (NTOK / TOK_PER_WG) // 512

#define CB_STRIDE 68                    // 64 + 4 pad -> conflict-free frag loads
#define ZT_STRIDE 68
#define SC_STRIDE 66

// output layout (floats), concatenated in reference return order
static constexpr size_t OFF_Z      = 0;                       // (B,L,DIM)   33554432
static constexpr size_t OFF_CRYST  = 33554432;                // (B,L,H)       524288
static constexpr size_t OFF_NEWLY  = 34078720;                // (B,L,H)       524288
static constexpr size_t OFF_CC     = 34603008;                // (B,L,H)       524288
static constexpr size_t OFF_FROZEN = 35127296;                // (B,L,H,DH)  33554432

__global__ __launch_bounds__(THREADS_PER_WG)
void crystallise_kernel(const float* __restrict__ z_cur,
                        const float* __restrict__ z_prev,
                        const float* __restrict__ codebook,
                        const float* __restrict__ frozen,
                        const int*   __restrict__ ccount,
                        const int*   __restrict__ cryst,
                        float* __restrict__ out)
{
    __shared__ float cb_s[M_ * CB_STRIDE];                    // head codebook, padded
    __shared__ float cbn_s[M_];                               // ||cb_m||^2
    __shared__ float zt_s[WAVES_PER_WG * 16 * ZT_STRIDE];     // z tiles
    __shared__ float sc_s[WAVES_PER_WG * 16 * SC_STRIDE];     // distance scores
    __shared__ int   fl_s[WAVES_PER_WG * 16];                 // per-token flags

    const int tid  = threadIdx.x;
    const int wave = tid >> 5;
    const int lane = tid & 31;
    const int lid  = lane & 15;
    const int half = lane >> 4;

    const int h       = blockIdx.x / WGS_PER_HEAD;
    const int tg      = blockIdx.x % WGS_PER_HEAD;
    const int tokBase = tg * TOK_PER_WG + wave * 16;          // this wave's 16 tokens

    // ---- stage head codebook into LDS (coalesced global reads) ----
    const float* cb_h = codebook + (size_t)h * M_ * DH_;
    for (int i = tid; i < M_ * DH_; i += THREADS_PER_WG) {
        int e = i >> 6;
        int d = i & 63;
        cb_s[e * CB_STRIDE + d] = cb_h[i];
    }
    __syncthreads();
    if (tid < M_) {
        float s = 0.f;
        #pragma unroll
        for (int d = 0; d < DH_; ++d) {
            float v = cb_s[tid * CB_STRIDE + d];
            s += v * v;
        }
        cbn_s[tid] = s;
    }

    // ---- load 16x64 z tile (coalesced float2/row), fuse velocity^2 ----
    float vel2 = 0.f;
    #pragma unroll
    for (int r = 0; r < 16; ++r) {
        size_t g = (size_t)(tokBase + r) * DIM_ + (size_t)h * DH_ + (size_t)(lane * 2);
        v2f zc = *(const v2f*)(z_cur  + g);
        v2f zp = *(const v2f*)(z_prev + g);
        *(v2f*)&zt_s[(wave * 16 + r) * ZT_STRIDE + lane * 2] = zc;
        float dx = zc.x - zp.x;
        float dy = zc.y - zp.y;
        float d  = dx * dx + dy * dy;
        #pragma unroll
        for (int off = 16; off >= 1; off >>= 1)
            d += __shfl_xor(d, off, 32);
        if (lane == r) vel2 = d;                              // lane r keeps token r's ||dz||^2
    }
    __syncthreads();                                          // cb_s/cbn_s + zt_s visible

    // ---- WMMA: dot[token 0..15][entry 0..63], f32 16x16x4, K=64 in 16 steps ----
    v8f acc0 = {}, acc1 = {}, acc2 = {}, acc3 = {};
    const float* zt_row = &zt_s[(wave * 16 + lid) * ZT_STRIDE];
    #pragma unroll
    for (int kk = 0; kk < 16; ++kk) {
        int col = kk * 4 + half * 2;                          // K pair per half-wave
        v2f a  = *(const v2f*)&zt_row[col];                   // A[m=lid][k,k+1]
        v2f b0 = *(const v2f*)&cb_s[(0 * 16 + lid) * CB_STRIDE + col]; // B[k][n]=cb[n][k]
        v2f b1 = *(const v2f*)&cb_s[(1 * 16 + lid) * CB_STRIDE + col];
        v2f b2 = *(const v2f*)&cb_s[(2 * 16 + lid) * CB_STRIDE + col];
        v2f b3 = *(const v2f*)&cb_s[(3 * 16 + lid) * CB_STRIDE + col];
        acc0 = __builtin_amdgcn_wmma_f32_16x16x4_f32(false, a, false, b0, (short)0, acc0, false, false);
        acc1 = __builtin_amdgcn_wmma_f32_16x16x4_f32(false, a, false, b1, (short)0, acc1, false, false);
        acc2 = __builtin_amdgcn_wmma_f32_16x16x4_f32(false, a, false, b2, (short)0, acc2, false, false);
        acc3 = __builtin_amdgcn_wmma_f32_16x16x4_f32(false, a, false, b3, (short)0, acc3, false, false);
    }

    // ---- scores = ||cb||^2 - 2*dot  (argmin-equivalent to full d2) ----
    float* sc_base = &sc_s[wave * 16 * SC_STRIDE];
    #pragma unroll
    for (int i = 0; i < 8; ++i) {
        int m = i + half * 8;                                 // C/D layout: VGPR i, halves
        sc_base[m * SC_STRIDE +  0 + lid] = cbn_s[ 0 + lid] - 2.f * acc0[i];
        sc_base[m * SC_STRIDE + 16 + lid] = cbn_s[16 + lid] - 2.f * acc1[i];
        sc_base[m * SC_STRIDE + 32 + lid] = cbn_s[32 + lid] - 2.f * acc2[i];
        sc_base[m * SC_STRIDE + 48 + lid] = cbn_s[48 + lid] - 2.f * acc3[i];
    }
    __syncthreads();

    // ---- per-token argmin (first-min tie-break) + crystallisation logic ----
    if (lane < 16) {
        int token = tokBase + lane;
        float best = 3.402823466e38f;
        int bidx = 0;
        for (int e = 0; e < M_; ++e) {
            float s = sc_base[lane * SC_STRIDE + e];
            if (s < best) { best = s; bidx = e; }
        }
        bool conv = vel2 < (0.01f * 0.01f);                   // ||dz|| < tau
        int  cco  = ccount[(size_t)token * H_ + h];
        int  ccn  = conv ? (cco + 1) : 0;
        bool cr   = cryst[(size_t)token * H_ + h] != 0;
        bool nw   = (ccn >= 2) && !cr;
        bool cn   = cr || nw;
        out[OFF_CRYST + (size_t)token * H_ + h] = cn ? 1.f : 0.f;
        out[OFF_NEWLY + (size_t)token * H_ + h] = nw ? 1.f : 0.f;
        out[OFF_CC    + (size_t)token * H_ + h] = (float)ccn;
        fl_s[wave * 16 + lane] = bidx | (nw ? 0x100 : 0) | (cn ? 0x200 : 0);
    }
    __syncthreads();

    // ---- cooperative coalesced writes: frozen_new + z_enforced ----
    #pragma unroll
    for (int r = 0; r < 16; ++r) {
        int  fl   = fl_s[wave * 16 + r];
        int  bidx = fl & 0xff;
        bool nw   = (fl & 0x100) != 0;
        bool cn   = (fl & 0x200) != 0;
        size_t g = (size_t)(tokBase + r) * DIM_ + (size_t)h * DH_ + (size_t)(lane * 2);
        v2f fv  = *(const v2f*)(frozen + g);
        v2f cbv = *(const v2f*)&cb_s[bidx * CB_STRIDE + lane * 2];
        v2f fn  = nw ? cbv : fv;
        v2f zc  = *(const v2f*)&zt_s[(wave * 16 + r) * ZT_STRIDE + lane * 2];
        v2f ze  = cn ? fn : zc;
        *(v2f*)(out + OFF_FROZEN + g) = fn;
        *(v2f*)(out + OFF_Z      + g) = ze;
    }
}

extern "C" void kernel_launch(void* const* d_in, const int* in_sizes, int n_in,
                              void* d_out, int out_size, void* d_ws, size_t ws_size,
                              hipStream_t stream) {
    (void)in_sizes; (void)n_in; (void)out_size; (void)d_ws; (void)ws_size;
    const float* z_cur  = (const float*)d_in[0];
    const float* z_prev = (const float*)d_in[1];
    const float* cb     = (const float*)d_in[2];
    const float* fv     = (const float*)d_in[3];
    const int*   cc     = (const int*)d_in[4];
    const int*   cr     = (const int*)d_in[5];
    float* out = (float*)d_out;

    dim3 grid(H_ * WGS_PER_HEAD);   // 8192 workgroups
    dim3 block(THREADS_PER_WG);     // 128 threads = 4 waves (wave32)
    crystallise_kernel<<<grid, block, 0, stream>>>(z_cur, z_prev, cb, fv, cc, cr, out);
}